// Decoder_17068200034466
// MI455X (gfx1250) — compile-verified
//
#include <hip/hip_runtime.h>
#include <hip/hip_bf16.h>

// ---------------------------------------------------------------------------
// CDNA5 (gfx1250) implementation.
// Activations live in workspace as (C x P) row-major, P = 8192 = b*256 + n.
// All conv1 ops are (O x C) @ (C x P) GEMMs done with V_WMMA_F32_16X16X4_F32
// (fp32 in/out -> preserves reference precision; we're bandwidth-bound anyway).
// ---------------------------------------------------------------------------

typedef __attribute__((ext_vector_type(2))) float v2f;
typedef __attribute__((ext_vector_type(8))) float v8f;

#define PPOS  8192   // 32 batches * 256 positions
#define NBAT  32
#define NPTS  33024  // 256 (pcd) + 32768 (partial)

__device__ __forceinline__ v8f wmma_f32_16x16x4(v2f a, v2f b, v8f c) {
  // 8 args: (neg_a, A, neg_b, B, c_mod, C, reuse_a, reuse_b)
  return __builtin_amdgcn_wmma_f32_16x16x4_f32(false, a, false, b, (short)0, c,
                                               false, false);
}

// ---------------------------------------------------------------------------
// Kernel 0: x1 = einsum('bi,iok->bok', f, ps_w) + ps_b, written as (C x P).
// GEMM view: Out[j, b] = sum_i psw[i*32768 + j] * feat[b*512 + i]
//   M = 32768 (j = o*256+k), N = 32 (b), K = 512.
// One wave per 16x16 tile; 4 waves per block; 4096 tiles -> 1024 blocks.
// ---------------------------------------------------------------------------
__global__ __launch_bounds__(128) void gemm_ps_kernel(
    const float* __restrict__ psw,   // (512, 32768)
    const float* __restrict__ feat,  // (32, 512)
    const float* __restrict__ psb,   // (128)
    float* __restrict__ X1)          // (128 x 8192)
{
  const int lane = threadIdx.x & 31;
  const int wid  = threadIdx.x >> 5;
  const int tile = blockIdx.x * 4 + wid;
  const int bt = tile & 1;          // N tile (2 of them)
  const int jt = tile >> 1;         // M tile (2048 of them)
  const int j0 = jt << 4;
  const int b0 = bt << 4;
  const int lm = lane & 15;
  const int kb = (lane >> 4) << 1;  // 0 or 2

  v8f acc = {};
  for (int k0 = 0; k0 < 512; k0 += 4) {
    // A[m][kk] = psw[(k0+kk)*32768 + j0+m]
    const float* ar = psw + (size_t)(k0 + kb) * 32768 + j0 + lm;
    v2f a; a.x = ar[0]; a.y = ar[32768];
    // stream hint for the 64MB weight tensor (emits global_prefetch_b8)
    __builtin_prefetch(ar + 4 * 32768, 0, 0);
    // B[kk][n] = feat[(b0+n)*512 + k0+kk]
    const float* br = feat + (b0 + lm) * 512 + k0 + kb;
    v2f b; b.x = br[0]; b.y = br[1];
    acc = wmma_f32_16x16x4(a, b, acc);
  }

  const int n = lane & 15;
  const int bb = b0 + n;
#pragma unroll
  for (int v = 0; v < 8; ++v) {
    const int m = ((lane >> 4) << 3) + v;
    const int j = j0 + m;
    const int o = j >> 8;
    const int k = j & 255;
    X1[o * PPOS + bb * 256 + k] = acc[v] + psb[o];
  }
}

// ---------------------------------------------------------------------------
// Kernel 1: fold the broadcast-feat half (columns 128..639) of m1_w1/m1_ws/
// m3_w1/m3_ws into per-(o,b) offsets (bias folded in too).
// T layout: [t1 | ts1 | t3 | ts3], each (128 x 32).
// ---------------------------------------------------------------------------
__global__ __launch_bounds__(256) void tvec_kernel(
    const float* __restrict__ feat,
    const float* __restrict__ w1_1, const float* __restrict__ b1_1,
    const float* __restrict__ ws_1, const float* __restrict__ bs_1,
    const float* __restrict__ w1_3, const float* __restrict__ b1_3,
    const float* __restrict__ ws_3, const float* __restrict__ bs_3,
    float* __restrict__ T)
{
  const int t = blockIdx.x * blockDim.x + threadIdx.x;  // 0..16383
  const int which = t >> 12;
  const int o = (t >> 5) & 127;
  const int b = t & 31;
  const float* W;
  const float* bias;
  if (which == 0)      { W = w1_1; bias = b1_1; }
  else if (which == 1) { W = ws_1; bias = bs_1; }
  else if (which == 2) { W = w1_3; bias = b1_3; }
  else                 { W = ws_3; bias = bs_3; }
  float acc = 0.f;
  const float* wr = W + o * 640 + 128;
  const float* fr = feat + b * 512;
  for (int c = 0; c < 512; ++c) acc += wr[c] * fr[c];
  T[t] = acc + bias[o];
}

// ---------------------------------------------------------------------------
// Generic conv1 GEMM: Y(OxP) = act( W(OxC,ldW) @ X(CxP) + tvec|bias + skip )
// O in {64,128}, C in {64,128}. One wave per 16x16 tile, 4 waves/block.
// Grid must be exactly (O/16)*(P/16)/4 blocks (full EXEC for WMMA).
// ---------------------------------------------------------------------------
__global__ __launch_bounds__(128) void conv_gemm_wmma(
    const float* __restrict__ W, int ldW,
    const float* __restrict__ X,
    float* __restrict__ Y,
    const float* __restrict__ tvec,   // (O x 32) or nullptr
    const float* __restrict__ bias,   // (O) or nullptr
    const float* __restrict__ skip,   // (O x P) or nullptr
    int O, int C, int relu)
{
  const int lane = threadIdx.x & 31;
  const int wid  = threadIdx.x >> 5;
  const int tilesM = O >> 4;                 // 4 or 8 (power of two)
  const int tile = blockIdx.x * 4 + wid;
  const int mt = tile & (tilesM - 1);
  const int nt = tile / tilesM;
  const int m0 = mt << 4;
  const int p0 = nt << 4;
  const int lm = lane & 15;
  const int kb = (lane >> 4) << 1;

  v8f acc = {};
  for (int k0 = 0; k0 < C; k0 += 4) {
    const float* wr = W + (m0 + lm) * ldW + k0 + kb;
    v2f a; a.x = wr[0]; a.y = wr[1];
    const float* xr = X + (k0 + kb) * PPOS + p0 + lm;
    v2f b; b.x = xr[0]; b.y = xr[PPOS];
    acc = wmma_f32_16x16x4(a, b, acc);
  }

  const int p = p0 + (lane & 15);
  const int bidx = p >> 8;
#pragma unroll
  for (int v = 0; v < 8; ++v) {
    const int o = m0 + ((lane >> 4) << 3) + v;
    float val = acc[v];
    if (tvec) val += tvec[o * 32 + bidx];
    if (bias) val += bias[o];
    if (skip) val += skip[o * PPOS + p];
    if (relu) val = fmaxf(val, 0.f);
    Y[o * PPOS + p] = val;
  }
}

// ---------------------------------------------------------------------------
// Head: completion = m4_w2(3x64) @ relu_hidden(64xP) + m4_b2, stored
// transposed as pcd (32, 256, 3) at the start of d_out.
// ---------------------------------------------------------------------------
__global__ __launch_bounds__(256) void final_conv_kernel(
    const float* __restrict__ H,    // (64 x 8192), already ReLU'd
    const float* __restrict__ w2,   // (3 x 64)
    const float* __restrict__ b2,   // (3)
    float* __restrict__ out)        // pcd (32,256,3)
{
  const int p = blockIdx.x * blockDim.x + threadIdx.x;
  float a0 = b2[0], a1 = b2[1], a2 = b2[2];
  for (int c = 0; c < 64; ++c) {
    const float h = H[c * PPOS + p];
    a0 += w2[c]       * h;
    a1 += w2[64 + c]  * h;
    a2 += w2[128 + c] * h;
  }
  const int b = p >> 8, n = p & 255;
  float* dst = out + (b * 256 + n) * 3;
  dst[0] = a0; dst[1] = a1; dst[2] = a2;
}

// ---------------------------------------------------------------------------
// Farthest point sampling: one workgroup per batch, 1024 threads, each
// thread caches 33 points + running min-dist entirely in VGPRs. Per step:
// pure VALU scan + shfl_xor wave argmax + LDS cross-wave argmax (2 barriers).
// Emits points[last] each step (matches jax scan ordering incl. idx 0 first).
// ---------------------------------------------------------------------------
__global__ __launch_bounds__(1024) void fps_kernel(
    const float* __restrict__ pcd,      // (32,256,3)  = d_out base
    const float* __restrict__ partial,  // (32,32768,3)
    float* __restrict__ p0)             // (32,512,3)  = d_out + 24576
{
  const int b = blockIdx.x;
  const int tid = threadIdx.x;
  const int lane = tid & 31;
  const int wid = tid >> 5;

  float px[33], py[33], pz[33], d[33];
#pragma unroll
  for (int k = 0; k < 33; ++k) {
    const int i = k * 1024 + tid;
    const bool valid = i < NPTS;
    float x = 0.f, y = 0.f, z = 0.f;
    if (valid) {
      const float* src = (i < 256) ? (pcd + (b * 256 + i) * 3)
                                   : (partial + ((size_t)b * 32768 + (i - 256)) * 3);
      x = src[0]; y = src[1]; z = src[2];
    }
    px[k] = x; py[k] = y; pz[k] = z;
    d[k] = valid ? 1e10f : -1.0f;   // invalid lanes can never win argmax
  }

  __shared__ float sbx, sby, sbz;
  __shared__ float swv[32];
  __shared__ int   swi[32];
  __shared__ int   sgi;

  if (tid == 0) {  // initial index = 0 -> pcd[b][0]
    const float* s0 = pcd + b * 256 * 3;
    sbx = s0[0]; sby = s0[1]; sbz = s0[2];
  }
  __syncthreads();

  for (int s = 0; s < 512; ++s) {
    const float lx = sbx, ly = sby, lz = sbz;
    if (tid == 0) {
      float* o = p0 + (b * 512 + s) * 3;
      o[0] = lx; o[1] = ly; o[2] = lz;
    }

    float bestv = -2.0f;
    int besti = 0;
#pragma unroll
    for (int k = 0; k < 33; ++k) {
      const float dx = px[k] - lx, dy = py[k] - ly, dz = pz[k] - lz;
      const float dist = dx * dx + dy * dy + dz * dz;
      const float nd = fminf(d[k], dist);
      d[k] = nd;
      if (nd > bestv) { bestv = nd; besti = k * 1024 + tid; }
    }
    // wave32 argmax (lowest index wins ties, like jnp.argmax)
#pragma unroll
    for (int off = 16; off; off >>= 1) {
      const float ov = __shfl_xor(bestv, off, 32);
      const int   oi = __shfl_xor(besti, off, 32);
      if (ov > bestv || (ov == bestv && oi < besti)) { bestv = ov; besti = oi; }
    }
    if (lane == 0) { swv[wid] = bestv; swi[wid] = besti; }
    __syncthreads();
    if (wid == 0) {
      float v = swv[lane];
      int   i = swi[lane];
#pragma unroll
      for (int off = 16; off; off >>= 1) {
        const float ov = __shfl_xor(v, off, 32);
        const int   oi = __shfl_xor(i, off, 32);
        if (ov > v || (ov == v && oi < i)) { v = ov; i = oi; }
      }
      if (lane == 0) {
        const int i2 = i;
        const float* src = (i2 < 256) ? (pcd + (b * 256 + i2) * 3)
                                      : (partial + ((size_t)b * 32768 + (i2 - 256)) * 3);
        sbx = src[0]; sby = src[1]; sbz = src[2];
        sgi = i2;
      }
    }
    __syncthreads();
  }
  (void)sgi;
}

// ---------------------------------------------------------------------------
extern "C" void kernel_launch(void* const* d_in, const int* in_sizes, int n_in,
                              void* d_out, int out_size, void* d_ws, size_t ws_size,
                              hipStream_t stream) {
  (void)in_sizes; (void)n_in; (void)out_size; (void)ws_size;
  const float* feat    = (const float*)d_in[0];
  const float* partial = (const float*)d_in[1];
  const float* ps_w    = (const float*)d_in[2];
  const float* ps_b    = (const float*)d_in[3];
  const float* m1_w1 = (const float*)d_in[4];  const float* m1_b1 = (const float*)d_in[5];
  const float* m1_w2 = (const float*)d_in[6];  const float* m1_b2 = (const float*)d_in[7];
  const float* m1_ws = (const float*)d_in[8];  const float* m1_bs = (const float*)d_in[9];
  const float* m2_w1 = (const float*)d_in[10]; const float* m2_b1 = (const float*)d_in[11];
  const float* m2_w2 = (const float*)d_in[12]; const float* m2_b2 = (const float*)d_in[13];
  const float* m2_ws = (const float*)d_in[14]; const float* m2_bs = (const float*)d_in[15];
  const float* m3_w1 = (const float*)d_in[16]; const float* m3_b1 = (const float*)d_in[17];
  const float* m3_w2 = (const float*)d_in[18]; const float* m3_b2 = (const float*)d_in[19];
  const float* m3_ws = (const float*)d_in[20]; const float* m3_bs = (const float*)d_in[21];
  const float* m4_w1 = (const float*)d_in[22]; const float* m4_b1 = (const float*)d_in[23];
  const float* m4_w2 = (const float*)d_in[24]; const float* m4_b2 = (const float*)d_in[25];
  (void)m3_b1; (void)m3_bs; (void)m1_b1; (void)m1_bs;

  // workspace: 3 ping-pong activation buffers (128 x 8192) + folded t-vectors
  float* A = (float*)d_ws;                  // current activation
  float* H = A + 128 * PPOS;                // hidden
  float* S = H + 128 * PPOS;                // shortcut
  float* T = S + 128 * PPOS;                // 4 * (128 x 32)

  float* pcd = (float*)d_out;               // (32,256,3)
  float* p0  = pcd + NBAT * 256 * 3;        // (32,512,3)

  // 1) point-splitting GEMM (+ps_b) -> A = x1 (128 x P)
  gemm_ps_kernel<<<1024, 128, 0, stream>>>(ps_w, feat, ps_b, A);
  // 2) fold broadcast-feat columns of m1/m3 w1,ws (+their biases) into T
  tvec_kernel<<<64, 256, 0, stream>>>(feat, m1_w1, m1_b1, m1_ws, m1_bs,
                                      m3_w1, m3_b1, m3_ws, m3_bs, T);
  // 3) mlp_res #1 (K folded 640 -> 128)
  conv_gemm_wmma<<<1024, 128, 0, stream>>>(m1_w1, 640, A, H, T,        nullptr, nullptr, 128, 128, 1);
  conv_gemm_wmma<<<1024, 128, 0, stream>>>(m1_ws, 640, A, S, T + 4096, nullptr, nullptr, 128, 128, 0);
  conv_gemm_wmma<<<1024, 128, 0, stream>>>(m1_w2, 128, H, A, nullptr,  m1_b2,  S,       128, 128, 0);
  // 4) mlp_res #2
  conv_gemm_wmma<<<512, 128, 0, stream>>>(m2_w1, 128, A, H, nullptr, m2_b1, nullptr,  64, 128, 1);
  conv_gemm_wmma<<<1024, 128, 0, stream>>>(m2_ws, 128, A, S, nullptr, m2_bs, nullptr, 128, 128, 0);
  conv_gemm_wmma<<<1024, 128, 0, stream>>>(m2_w2,  64, H, A, nullptr, m2_b2, S,       128,  64, 0);
  // 5) mlp_res #3 (K folded 640 -> 128)
  conv_gemm_wmma<<<1024, 128, 0, stream>>>(m3_w1, 640, A, H, T + 8192,  nullptr, nullptr, 128, 128, 1);
  conv_gemm_wmma<<<1024, 128, 0, stream>>>(m3_ws, 640, A, S, T + 12288, nullptr, nullptr, 128, 128, 0);
  conv_gemm_wmma<<<1024, 128, 0, stream>>>(m3_w2, 128, H, A, nullptr,   m3_b2,  S,       128, 128, 0);
  // 6) head: 128 -> 64 (relu) -> 3, transpose-store pcd
  conv_gemm_wmma<<<512, 128, 0, stream>>>(m4_w1, 128, A, H, nullptr, m4_b1, nullptr, 64, 128, 1);
  final_conv_kernel<<<32, 256, 0, stream>>>(H, m4_w2, m4_b2, pcd);
  // 7) farthest point sampling over [pcd ; partial], 512 samples
  fps_kernel<<<32, 1024, 0, stream>>>(pcd, partial, p0);
}